// DTW_1d_dim_18098992185357
// MI455X (gfx1250) — compile-verified
//
#include <hip/hip_runtime.h>

typedef float v2f __attribute__((ext_vector_type(2)));
typedef float v8f __attribute__((ext_vector_type(8)));

#define NC   64
#define NL   4096
#define NDIM 5
#define NB   8
#define LOUT (NL - 3)      // 4093
#define NTILE (NL / 16)    // 256

// ---------------------------------------------------------------------------
// Kernel 0: squared-norm accumulations over channels.
//   Sx2[b][m] = sum_c x[b,c,m]^2              (B*L)
//   SD2[d][m] = sum_c DWT_all[c,m,d]^2        (DIM*L)
// ---------------------------------------------------------------------------
__global__ void norms_kernel(const float* __restrict__ x,
                             const float* __restrict__ DWT,
                             float* __restrict__ Sx2,
                             float* __restrict__ SD2) {
    int tid = blockIdx.x * blockDim.x + threadIdx.x;
    if (tid < NB * NL) {
        int b = tid / NL, m = tid % NL;
        float s = 0.f;
        #pragma unroll 4
        for (int c = 0; c < NC; ++c) {
            float v = x[((size_t)b * NC + c) * NL + m];
            s += v * v;
        }
        Sx2[tid] = s;
    } else if (tid < (NB + NDIM) * NL) {
        int r = tid - NB * NL;
        int d = r / NL, m = r % NL;
        float s = 0.f;
        #pragma unroll 4
        for (int c = 0; c < NC; ++c) {
            float v = DWT[((size_t)c * NL + m) * NDIM + d];
            s += v * v;
        }
        SD2[r] = s;
    }
}

// ---------------------------------------------------------------------------
// Kernel 1: banded cross-correlation via V_WMMA_F32_16X16X4_F32.
//   G[b][d][o][m] = sum_c x[b,c,m] * DWT_all[c, m+o-2, d],  o in [0,5) (offset o-2)
// One wave (32 threads) per (b, 16-row tile). All 5 dims per wave:
//   10 accumulator tiles (5 dims x 2 column tiles), K=64 via 16 chained WMMAs.
// Column tile bases M0-8 and M0+8 cover the needed band n in [M0-2, M0+18).
// Edge columns are CLAMPED (not zeroed): clamped columns only produce P-tile
// entries whose extracted G values are never read by the combine kernel, so
// no EXEC-diverging guards are needed and WMMA sees all-ones EXEC throughout.
// ---------------------------------------------------------------------------
__global__ __launch_bounds__(32) void banded_wmma_kernel(
        const float* __restrict__ x,
        const float* __restrict__ DWT,
        float* __restrict__ G) {
    const int blk  = blockIdx.x;
    const int tile = blk % NTILE;
    const int b    = blk / NTILE;
    const int M0   = tile * 16;

    const int lane = threadIdx.x;      // 0..31
    const int row  = lane & 15;        // A-row / B-col for this lane
    const int half = lane >> 4;        // 0: K pair {0,1}; 1: K pair {2,3}
    const int kb   = half * 2;

    v8f acc[NDIM][2];
    #pragma unroll
    for (int d = 0; d < NDIM; ++d) {
        #pragma unroll
        for (int t = 0; t < 2; ++t)
            acc[d][t] = (v8f){0.f, 0.f, 0.f, 0.f, 0.f, 0.f, 0.f, 0.f};
    }

    const float* xb = x + (size_t)b * NC * NL;

    const int col = row;
    int n0 = M0 - 8 + col;             // column tile 0 base (may underflow at M0=0)
    int n1 = M0 + 8 + col;             // column tile 1 base (may overflow at M0=4080)
    n0 = (n0 < 0) ? 0 : n0;            // clamp: affected columns are never consumed
    n1 = (n1 > NL - 1) ? (NL - 1) : n1;

    for (int s = 0; s < 16; ++s) {
        const int c0 = 4 * s + kb;
        // A operand (16x4 f32): lane row, K = c0, c0+1
        v2f a;
        a.x = xb[(size_t)c0       * NL + M0 + row];
        a.y = xb[(size_t)(c0 + 1) * NL + M0 + row];

        const size_t r0 = (size_t)c0 * NL;
        const size_t r1 = (size_t)(c0 + 1) * NL;

        #pragma unroll
        for (int d = 0; d < NDIM; ++d) {
            v2f b0, b1;
            b0.x = DWT[(r0 + n0) * NDIM + d];
            b0.y = DWT[(r1 + n0) * NDIM + d];
            b1.x = DWT[(r0 + n1) * NDIM + d];
            b1.y = DWT[(r1 + n1) * NDIM + d];
            acc[d][0] = __builtin_amdgcn_wmma_f32_16x16x4_f32(
                false, a, false, b0, (short)0, acc[d][0], false, false);
            acc[d][1] = __builtin_amdgcn_wmma_f32_16x16x4_f32(
                false, a, false, b1, (short)0, acc[d][1], false, false);
        }
    }

    // Spill the 16x32 P tiles (per dim) to LDS, then extract the 5 diagonals.
    __shared__ float lds[NDIM][16][32];
    #pragma unroll
    for (int d = 0; d < NDIM; ++d) {
        #pragma unroll
        for (int v = 0; v < 8; ++v) {
            int r = v + 8 * half;                  // C/D layout: M = v + 8*(lane>>4)
            lds[d][r][col]      = acc[d][0][v];    // n = M0-8+col
            lds[d][r][col + 16] = acc[d][1][v];    // n = M0+8+col
        }
    }
    __syncthreads();

    // G[b][d][o][M0+r] = P[M0+r, M0+r+(o-2)] = lds[d][r][r + (o-2) + 8]
    for (int k = lane; k < NDIM * 5 * 16; k += 32) {
        int d  = k / 80;
        int rr = k % 80;
        int o  = rr / 16;
        int r  = rr % 16;
        G[(((size_t)b * NDIM + d) * 5 + o) * NL + M0 + r] = lds[d][r][r + o + 6];
    }
}

// ---------------------------------------------------------------------------
// Kernel 2: reconstruct the 14 t_ij terms, 9 distances, softmax, weighted sum.
//   t_ij = Sx2[b,l+i] - 2*G_{j-i}[b,d,l+i] + SD2[d,l+j]
//   out[b][d][l] = sum_k d_k * softmax(-d)_k
// ---------------------------------------------------------------------------
__global__ void combine_kernel(const float* __restrict__ Sx2,
                               const float* __restrict__ SD2,
                               const float* __restrict__ G,
                               float* __restrict__ out) {
    int tid = blockIdx.x * blockDim.x + threadIdx.x;
    if (tid >= NDIM * NB * LOUT) return;
    int l = tid % LOUT;
    int r = tid / LOUT;
    int d = r % NDIM;
    int b = r / NDIM;

    const float* sx = Sx2 + (size_t)b * NL + l;
    const float* sd = SD2 + (size_t)d * NL + l;
    const size_t gb = ((size_t)b * NDIM + d) * 5 * NL;
    const float* g0 = G + gb + 0 * (size_t)NL + l;   // offset -2
    const float* g1 = G + gb + 1 * (size_t)NL + l;   // offset -1
    const float* g2 = G + gb + 2 * (size_t)NL + l;   // offset  0
    const float* g3 = G + gb + 3 * (size_t)NL + l;   // offset +1
    const float* g4 = G + gb + 4 * (size_t)NL + l;   // offset +2

    float sx0 = sx[0], sx1 = sx[1], sx2 = sx[2], sx3 = sx[3];
    float sd0 = sd[0], sd1 = sd[1], sd2 = sd[2], sd3 = sd[3];

    float t00 = sx0 - 2.f * g2[0] + sd0;
    float t11 = sx1 - 2.f * g2[1] + sd1;
    float t22 = sx2 - 2.f * g2[2] + sd2;
    float t33 = sx3 - 2.f * g2[3] + sd3;
    float t01 = sx0 - 2.f * g3[0] + sd1;
    float t12 = sx1 - 2.f * g3[1] + sd2;
    float t23 = sx2 - 2.f * g3[2] + sd3;
    float t02 = sx0 - 2.f * g4[0] + sd2;
    float t13 = sx1 - 2.f * g4[1] + sd3;
    float t10 = sx1 - 2.f * g1[1] + sd0;
    float t21 = sx2 - 2.f * g1[2] + sd1;
    float t32 = sx3 - 2.f * g1[3] + sd2;
    float t20 = sx2 - 2.f * g0[2] + sd0;
    float t31 = sx3 - 2.f * g0[3] + sd1;

    float v0 = t00 + t11 + t22 + t33;                         // d1
    float v1 = t00 + t01 + t12 + t23 + t33;                   // d2
    float v2 = t00 + t01 + t02 + t13 + t23 + t33;             // d3
    float v3 = t00 + t11 + t12 + t23 + t33;                   // d4
    float v4 = t00 + t01 + t12 + t22 + t33;                   // d5
    float v5 = t00 + t10 + t21 + t32 + t33;                   // d2b
    float v6 = t00 + t10 + t20 + t31 + t32 + t33;             // d3b
    float v7 = t00 + t11 + t21 + t32 + t33;                   // d4b
    float v8 = t00 + t10 + t21 + t22 + t33;                   // d5b

    float mn = fminf(v0, v1);
    mn = fminf(mn, v2); mn = fminf(mn, v3); mn = fminf(mn, v4);
    mn = fminf(mn, v5); mn = fminf(mn, v6); mn = fminf(mn, v7);
    mn = fminf(mn, v8);

    float w, wsum = 0.f, dsum = 0.f;
    w = __expf(mn - v0); wsum += w; dsum += v0 * w;
    w = __expf(mn - v1); wsum += w; dsum += v1 * w;
    w = __expf(mn - v2); wsum += w; dsum += v2 * w;
    w = __expf(mn - v3); wsum += w; dsum += v3 * w;
    w = __expf(mn - v4); wsum += w; dsum += v4 * w;
    w = __expf(mn - v5); wsum += w; dsum += v5 * w;
    w = __expf(mn - v6); wsum += w; dsum += v6 * w;
    w = __expf(mn - v7); wsum += w; dsum += v7 * w;
    w = __expf(mn - v8); wsum += w; dsum += v8 * w;

    out[((size_t)b * NDIM + d) * LOUT + l] = dsum / wsum;
}

// ---------------------------------------------------------------------------
extern "C" void kernel_launch(void* const* d_in, const int* in_sizes, int n_in,
                              void* d_out, int out_size, void* d_ws, size_t ws_size,
                              hipStream_t stream) {
    const float* x   = (const float*)d_in[0];   // (B, C, L) f32
    const float* DWT = (const float*)d_in[1];   // (C, L, DIM) f32
    float* out = (float*)d_out;                 // (B, DIM, LOUT) f32

    // Workspace layout (floats): Sx2 [B*L] | SD2 [DIM*L] | G [B*DIM*5*L]
    float* Sx2 = (float*)d_ws;
    float* SD2 = Sx2 + (size_t)NB * NL;
    float* G   = SD2 + (size_t)NDIM * NL;

    int normThreads = (NB + NDIM) * NL;
    norms_kernel<<<(normThreads + 255) / 256, 256, 0, stream>>>(x, DWT, Sx2, SD2);

    banded_wmma_kernel<<<NB * NTILE, 32, 0, stream>>>(x, DWT, G);

    int combThreads = NDIM * NB * LOUT;
    combine_kernel<<<(combThreads + 255) / 256, 256, 0, stream>>>(Sx2, SD2, G, out);
}